// FeatureGradFetcher_69930657513596
// MI455X (gfx1250) — compile-verified
//
#include <hip/hip_runtime.h>

// Problem constants from the reference.
#define Bk 2
#define Vk 5
#define Ck 16
#define Hk 512
#define Wk 640
#define Nk 131072

typedef float f32x2 __attribute__((ext_vector_type(2)));

struct Samp {
    int   o00, o01, o10, o11;     // precomputed y*W+x offsets (clamped)
    float w00, w01, w10, w11;     // bilinear weights pre-multiplied by validity
};

__device__ __forceinline__ Samp make_samp(float gx, float gy) {
    Samp s;
    // ix = ((gx+1)*W - 1)*0.5 per reference
    float ix = ((gx + 1.0f) * (float)Wk - 1.0f) * 0.5f;
    float iy = ((gy + 1.0f) * (float)Hk - 1.0f) * 0.5f;
    float x0f = floorf(ix);
    float y0f = floorf(iy);
    float wx1 = ix - x0f, wy1 = iy - y0f;
    float wx0 = 1.0f - wx1, wy0 = 1.0f - wy1;
    float x1f = x0f + 1.0f, y1f = y0f + 1.0f;
    // validity masks (match reference: (xc>=0)&(xc<=W-1) on the float coord)
    float vx0 = (x0f >= 0.0f && x0f <= (float)(Wk - 1)) ? 1.0f : 0.0f;
    float vx1 = (x1f >= 0.0f && x1f <= (float)(Wk - 1)) ? 1.0f : 0.0f;
    float vy0 = (y0f >= 0.0f && y0f <= (float)(Hk - 1)) ? 1.0f : 0.0f;
    float vy1 = (y1f >= 0.0f && y1f <= (float)(Hk - 1)) ? 1.0f : 0.0f;
    // clamped integer coords (v_cvt saturates; clamp again for safety)
    int x0 = min(max((int)x0f, 0), Wk - 1);
    int x1 = min(max((int)x1f, 0), Wk - 1);
    int y0 = min(max((int)y0f, 0), Hk - 1);
    int y1 = min(max((int)y1f, 0), Hk - 1);
    s.o00 = y0 * Wk + x0;
    s.o01 = y0 * Wk + x1;
    s.o10 = y1 * Wk + x0;
    s.o11 = y1 * Wk + x1;
    s.w00 = wx0 * wy0 * vx0 * vy0;
    s.w01 = wx1 * wy0 * vx1 * vy0;
    s.w10 = wx0 * wy1 * vx0 * vy1;
    s.w11 = wx1 * wy1 * vx1 * vy1;
    return s;
}

__global__ __launch_bounds__(256) void feature_grad_fetch_kernel(
    const float* __restrict__ fm,     // (B*V, C, H, W)
    const float* __restrict__ pts,    // (B, 3, N)
    const float* __restrict__ Kmat,   // (B*V, 3, 3)
    const float* __restrict__ Emat,   // (B*V, 3, 4)
    float* __restrict__ f_out,        // (B*V, C, N)
    f32x2* __restrict__ g_out)        // (B*V, C, N) pairs {gx, gy}
{
    const int n  = blockIdx.x * blockDim.x + threadIdx.x;
    const int bv = blockIdx.y;                 // uniform per block -> scalar K/E
    if (n >= Nk) return;
    const int b = bv / Vk;

    // --- point load (read-once stream: non-temporal, don't pollute L2) ---
    const float px = __builtin_nontemporal_load(&pts[((size_t)b * 3 + 0) * Nk + n]);
    const float py = __builtin_nontemporal_load(&pts[((size_t)b * 3 + 1) * Nk + n]);
    const float pz = __builtin_nontemporal_load(&pts[((size_t)b * 3 + 2) * Nk + n]);

    // --- extrinsic transform (uniform matrix; compiler scalarizes loads) ---
    const float* E = Emat + (size_t)bv * 12;
    const float* K = Kmat + (size_t)bv * 9;
    float xx = E[0] * px + E[1] * py + E[2]  * pz + E[3];
    float yy = E[4] * px + E[5] * py + E[6]  * pz + E[7];
    float zz = E[8] * px + E[9] * py + E[10] * pz + E[11];

    // --- projection: uv = [x/z, y/z, 1] . K^T ---
    float nx = xx / zz;
    float ny = yy / zz;
    float u = K[0] * nx + K[1] * ny + K[2];
    float v = K[3] * nx + K[4] * ny + K[5];

    const float dx = 2.0f / (float)(Wk - 1);
    const float dy = 2.0f / (float)(Hk - 1);
    float gx = (u - 0.5f) / (float)(Wk - 1) * 2.0f - 1.0f;
    float gy = (v - 0.5f) / (float)(Hk - 1) * 2.0f - 1.0f;

    // --- precompute all five bilinear footprints ONCE, reuse for 16 channels ---
    Samp sC = make_samp(gx,      gy);
    Samp sL = make_samp(gx - dx, gy);
    Samp sR = make_samp(gx + dx, gy);
    Samp sT = make_samp(gx,      gy - dy);
    Samp sB = make_samp(gx,      gy + dy);

    const size_t HW    = (size_t)Hk * Wk;
    const float* plane = fm + (size_t)bv * Ck * HW;
    const size_t obase = (size_t)bv * Ck * Nk + (size_t)n;

#pragma unroll 2
    for (int c = 0; c < Ck; ++c) {
        const float* p = plane + (size_t)c * HW;

        // 20 gathers per channel, FMA-chained; all offsets/weights in registers.
        // These are the only accesses with reuse -> default RT policy keeps
        // the 21MB-per-view feature planes resident in the 192MB L2.
        float fc = sC.w00 * p[sC.o00];
        fc = fmaf(sC.w01, p[sC.o01], fc);
        fc = fmaf(sC.w10, p[sC.o10], fc);
        fc = fmaf(sC.w11, p[sC.o11], fc);

        float fl = sL.w00 * p[sL.o00];
        fl = fmaf(sL.w01, p[sL.o01], fl);
        fl = fmaf(sL.w10, p[sL.o10], fl);
        fl = fmaf(sL.w11, p[sL.o11], fl);

        float fr = sR.w00 * p[sR.o00];
        fr = fmaf(sR.w01, p[sR.o01], fr);
        fr = fmaf(sR.w10, p[sR.o10], fr);
        fr = fmaf(sR.w11, p[sR.o11], fr);

        float ft = sT.w00 * p[sT.o00];
        ft = fmaf(sT.w01, p[sT.o01], ft);
        ft = fmaf(sT.w10, p[sT.o10], ft);
        ft = fmaf(sT.w11, p[sT.o11], ft);

        float fb = sB.w00 * p[sB.o00];
        fb = fmaf(sB.w01, p[sB.o01], fb);
        fb = fmaf(sB.w10, p[sB.o10], fb);
        fb = fmaf(sB.w11, p[sB.o11], fb);

        // Output is written once and never re-read by this kernel:
        // non-temporal stores (TH=NT) avoid evicting feature-map lines from L2.
        __builtin_nontemporal_store(fc, &f_out[obase + (size_t)c * Nk]);
        f32x2 g;
        g.x = 0.5f * (fr - fl);
        g.y = 0.5f * (fb - ft);
        __builtin_nontemporal_store(g, &g_out[obase + (size_t)c * Nk]);
    }
}

extern "C" void kernel_launch(void* const* d_in, const int* in_sizes, int n_in,
                              void* d_out, int out_size, void* d_ws, size_t ws_size,
                              hipStream_t stream) {
    (void)in_sizes; (void)n_in; (void)out_size; (void)d_ws; (void)ws_size;

    const float* fm   = (const float*)d_in[0];   // feature_maps (B,V,C,H,W)
    const float* pts  = (const float*)d_in[1];   // pts (B,3,N)
    const float* Kms  = (const float*)d_in[2];   // cam_intrinsics (B,V,3,3)
    const float* Ems  = (const float*)d_in[3];   // cam_extrinsics (B,V,3,4)

    float* f_out = (float*)d_out;                                        // (B,V,C,N)
    f32x2* g_out = (f32x2*)((float*)d_out + (size_t)Bk * Vk * Ck * Nk);  // (B,V,C,N,2)

    dim3 block(256);
    dim3 grid((Nk + 255) / 256, Bk * Vk);
    feature_grad_fetch_kernel<<<grid, block, 0, stream>>>(fm, pts, Kms, Ems, f_out, g_out);
}